// EncodeProcessDecode_47218870453045
// MI455X (gfx1250) — compile-verified
//
#include <hip/hip_runtime.h>
#include <hip/hip_bf16.h>

// EncodeProcessDecode GNN for MI455X (gfx1250, wave32).
// All dense 64-wide MLP layers run on V_WMMA_F32_16X16X4_F32 (full fp32).
// One wave owns a 16-row x 64-out tile; a 64-thread block owns 32 rows.
// Ragged-K layers use LDS zero-padded weights so the WMMA loop is branch-free.

#define NN 50000
#define NE 800000
#define S_ACT 68   // LDS activation stride (floats), 68 % 64 == 4 -> conflict-free

typedef __attribute__((ext_vector_type(2))) float v2f;
typedef __attribute__((ext_vector_type(8))) float v8f;

// ---------------------------------------------------------------------------
// One MLP layer (K4 -> 64) for a wave's 16 rows via f32 WMMA.
// K4 must be a multiple of 4; A rows and W must be readable for all k < K4
// (zero-padded where the logical K is smaller).
// A (16x4 f32): lanes 0-15 row=lane K={kk,kk+1}; lanes 16-31 row=lane-16 K={kk+2,kk+3}
// B (4x16 f32): lane holds column n = lane%16, rows k0/k0+1 in the two VGPRs
// C/D (16x16):  vgpr i -> row hi*8+i, col nt*16 + lane%16
// ---------------------------------------------------------------------------
__device__ __forceinline__ void wmma_layer(const float* __restrict__ W,   // K4 x 64 row-major
                                           const float* __restrict__ Bv,  // 64 (bias)
                                           const float* Arow,             // lane's row
                                           int K4, v8f acc[4], int nlane, int hi)
{
#pragma unroll
  for (int nt = 0; nt < 4; ++nt) {
    float bval = Bv[nt * 16 + nlane];
    v8f t;
#pragma unroll
    for (int i = 0; i < 8; ++i) t[i] = bval;
    acc[nt] = t;
  }
  for (int kk = 0; kk < K4; kk += 4) {
    int k0 = kk + 2 * hi;
    v2f a;
    a.x = Arow[k0];
    a.y = Arow[k0 + 1];
    const float* wp = W + k0 * 64 + nlane;
#pragma unroll
    for (int nt = 0; nt < 4; ++nt) {
      v2f b;
      b.x = wp[nt * 16];
      b.y = wp[64 + nt * 16];
      acc[nt] = __builtin_amdgcn_wmma_f32_16x16x4_f32(false, a, false, b,
                                                      (short)0, acc[nt],
                                                      false, false);
    }
  }
}

__device__ __forceinline__ void store_acc(float* actw /* wave's 16-row base */,
                                          const v8f acc[4], int nlane, int hi,
                                          bool relu)
{
#pragma unroll
  for (int nt = 0; nt < 4; ++nt) {
#pragma unroll
    for (int i = 0; i < 8; ++i) {
      float v = acc[nt][i];
      if (relu) v = fmaxf(v, 0.0f);
      actw[(hi * 8 + i) * S_ACT + nt * 16 + nlane] = v;
    }
  }
}

// layers 2 (64->64, relu) + 3 (64->64) in-place in act[], then LN stats.
__device__ __forceinline__ void hidden_tail_ln(float* act, float* mrow, float* rrow,
                                               const float* W1, const float* B1,
                                               const float* W2, const float* B2)
{
  int tid = threadIdx.x;
  int wv = tid >> 5, lane = tid & 31;
  int nlane = lane & 15, hi = lane >> 4;
  const float* Arow = act + (wv * 16 + nlane) * S_ACT;
  float* actw = act + (wv * 16) * S_ACT;
  v8f acc[4];
  wmma_layer(W1, B1, Arow, 64, acc, nlane, hi);
  store_acc(actw, acc, nlane, hi, true);
  wmma_layer(W2, B2, Arow, 64, acc, nlane, hi);
  store_acc(actw, acc, nlane, hi, false);
  __syncthreads();
  int row = tid >> 1, half = tid & 1;
  const float* ar = act + row * S_ACT + half * 32;
  float s = 0.0f, ss = 0.0f;
#pragma unroll
  for (int c = 0; c < 32; ++c) { float v = ar[c]; s += v; ss += v * v; }
  s += __shfl_xor(s, 1);
  ss += __shfl_xor(ss, 1);
  float m = s * (1.0f / 64.0f);
  float var = ss * (1.0f / 64.0f) - m * m;
  float rstd = rsqrtf(var + 1e-5f);
  if (half == 0) { mrow[row] = m; rrow[row] = rstd; }
  __syncthreads();
}

// ---------------------------------------------------------------------------
// Utility kernels
// ---------------------------------------------------------------------------
__global__ void zero_kernel(float* __restrict__ p, int n)
{
  int i = blockIdx.x * blockDim.x + threadIdx.x;
  if (i < n) p[i] = 0.0f;
}

__device__ __forceinline__ float wave_sum(float v)
{
#pragma unroll
  for (int m = 16; m >= 1; m >>= 1) v += __shfl_xor(v, m);
  return v;
}

// acc[0..4] = col sums (x0,x1,a0,a1,dist), acc[5..9] = col sum-of-squares
__global__ void node_stats_kernel(const float* __restrict__ x, float* __restrict__ acc)
{
  int i = blockIdx.x * blockDim.x + threadIdx.x;
  float v0 = 0.0f, v1 = 0.0f;
  if (i < NN) { v0 = x[2 * i]; v1 = x[2 * i + 1]; }
  float s0 = wave_sum(v0), s1 = wave_sum(v1);
  float q0 = wave_sum(v0 * v0), q1 = wave_sum(v1 * v1);
  if ((threadIdx.x & 31) == 0) {
    atomicAdd(&acc[0], s0); atomicAdd(&acc[1], s1);
    atomicAdd(&acc[5], q0); atomicAdd(&acc[6], q1);
  }
}

__global__ void edge_stats_kernel(const float* __restrict__ ea, float* __restrict__ acc)
{
  int e = blockIdx.x * blockDim.x + threadIdx.x;
  float a0 = 0.0f, a1 = 0.0f, d = 0.0f;
  if (e < NE) { a0 = ea[2 * e]; a1 = ea[2 * e + 1]; d = sqrtf(a0 * a0 + a1 * a1); }
  float s0 = wave_sum(a0), s1 = wave_sum(a1), s2 = wave_sum(d);
  float q0 = wave_sum(a0 * a0), q1 = wave_sum(a1 * a1), q2 = wave_sum(a0 * a0 + a1 * a1);
  if ((threadIdx.x & 31) == 0) {
    atomicAdd(&acc[2], s0); atomicAdd(&acc[3], s1); atomicAdd(&acc[4], s2);
    atomicAdd(&acc[7], q0); atomicAdd(&acc[8], q1); atomicAdd(&acc[9], q2);
  }
}

// norm[0..4] = means, norm[8..12] = 1/std  (cols: x0,x1,a0,a1,dist)
__global__ void finalize_stats_kernel(const float* __restrict__ acc, float* __restrict__ norm)
{
  int i = threadIdx.x;
  if (i < 5) {
    float cnt = (i < 2) ? (float)NN : (float)NE;
    float m = acc[i] / cnt;
    float ex2 = acc[5 + i] / cnt;
    float sd = fmaxf(sqrtf(fmaxf(ex2 - m * m, 0.0f)), 1e-8f);
    norm[i] = m;
    norm[8 + i] = 1.0f / sd;
  }
}

// ---------------------------------------------------------------------------
// Light edge MLP (5->32->32->2) + lc scatter (pure VALU, ~1 GFLOP total)
// ---------------------------------------------------------------------------
__global__ void light_edge_kernel(const float* __restrict__ x, const int* __restrict__ ei,
                                  const float* __restrict__ ea,
                                  const float* __restrict__ w0, const float* __restrict__ b0,
                                  const float* __restrict__ w1, const float* __restrict__ b1,
                                  const float* __restrict__ w2, const float* __restrict__ b2,
                                  float* __restrict__ lc)
{
  __shared__ float sw[1314];  // w0(160) b0(32) w1(1024) b1(32) w2(64) b2(2)
  for (int i = threadIdx.x; i < 1314; i += 256) {
    float v;
    if (i < 160)       v = w0[i];
    else if (i < 192)  v = b0[i - 160];
    else if (i < 1216) v = w1[i - 192];
    else if (i < 1248) v = b1[i - 1216];
    else if (i < 1312) v = w2[i - 1248];
    else               v = b2[i - 1312];
    sw[i] = v;
  }
  __syncthreads();
  int e = blockIdx.x * blockDim.x + threadIdx.x;
  if (e >= NE) return;
  int r = ei[e], c = ei[NE + e];
  float a0 = ea[2 * e], a1 = ea[2 * e + 1];
  float dist = sqrtf(a0 * a0 + a1 * a1);
  float fj = x[2 * r], fi = x[2 * c], degree = x[2 * c + 1];
  float in[5] = { fj - fi, degree, a0, a1, dist };
  float h1[32];
#pragma unroll
  for (int o = 0; o < 32; ++o) {
    float s = sw[160 + o];
#pragma unroll
    for (int k = 0; k < 5; ++k) s += in[k] * sw[k * 32 + o];
    h1[o] = fmaxf(s, 0.0f);
  }
  float h2[32];
  for (int o = 0; o < 32; ++o) {
    float s = sw[1216 + o];
#pragma unroll
    for (int k = 0; k < 32; ++k) s += h1[k] * sw[192 + k * 32 + o];
    h2[o] = fmaxf(s, 0.0f);
  }
  float o0 = sw[1312], o1 = sw[1313];
#pragma unroll
  for (int k = 0; k < 32; ++k) {
    o0 += h2[k] * sw[1248 + 2 * k];
    o1 += h2[k] * sw[1248 + 2 * k + 1];
  }
  float df = fj - fi;
  float inv = 1.0f / (dist * dist);
  atomicAdd(&lc[2 * c],     df * inv * a0 * o0);
  atomicAdd(&lc[2 * c + 1], df * inv * a1 * o1);
}

// ---------------------------------------------------------------------------
// Node encoder: xn (K=2, padded to 4) -> 64 -> 64 -> 64 + LN -> xl
// ---------------------------------------------------------------------------
__global__ void __launch_bounds__(64)
enc_node_kernel(const float* __restrict__ x, const float* __restrict__ norm,
                const float* W0, const float* B0, const float* W1, const float* B1,
                const float* W2, const float* B2,
                const float* lng, const float* lnb, float* __restrict__ xl)
{
  __shared__ float sw0[4 * 64];     // W0 zero-padded K=2 -> 4
  __shared__ float inbuf[32 * 8];
  __shared__ float act[32 * S_ACT];
  __shared__ float mrow[32], rrow[32];
  int n0 = blockIdx.x * 32;
  int tid = threadIdx.x;
  for (int i = tid; i < 4 * 64; i += 64) {
    int k = i >> 6;
    sw0[i] = (k < 2) ? W0[i] : 0.0f;
  }
  for (int idx = tid; idx < 32 * 4; idx += 64) {
    int r = idx >> 2, c = idx & 3;
    int n = n0 + r;
    float v = 0.0f;
    if (n < NN && c < 2) v = (x[2 * n + c] - norm[c]) * norm[8 + c];
    inbuf[r * 8 + c] = v;
  }
  __syncthreads();
  int wv = tid >> 5, lane = tid & 31, nlane = lane & 15, hi = lane >> 4;
  v8f acc[4];
  wmma_layer(sw0, B0, inbuf + (wv * 16 + nlane) * 8, 4, acc, nlane, hi);
  store_acc(act + (wv * 16) * S_ACT, acc, nlane, hi, true);
  hidden_tail_ln(act, mrow, rrow, W1, B1, W2, B2);
  for (int idx = tid; idx < 32 * 64; idx += 64) {
    int r = idx >> 6, c = idx & 63;
    int n = n0 + r;
    if (n < NN)
      xl[n * 64 + c] = (act[r * S_ACT + c] - mrow[r]) * rrow[r] * lng[c] + lnb[c];
  }
}

// ---------------------------------------------------------------------------
// Edge encoder: ean (K=3, padded to 4) -> 64 -> 64 -> 64 + LN -> eal
// ---------------------------------------------------------------------------
__global__ void __launch_bounds__(64)
enc_edge_kernel(const float* __restrict__ ea, const float* __restrict__ norm,
                const float* W0, const float* B0, const float* W1, const float* B1,
                const float* W2, const float* B2,
                const float* lng, const float* lnb, float* __restrict__ eal)
{
  __shared__ float sw0[4 * 64];     // W0 zero-padded K=3 -> 4
  __shared__ float inbuf[32 * 8];
  __shared__ float act[32 * S_ACT];
  __shared__ float mrow[32], rrow[32];
  int e0 = blockIdx.x * 32;
  int tid = threadIdx.x;
  for (int i = tid; i < 4 * 64; i += 64) {
    int k = i >> 6;
    sw0[i] = (k < 3) ? W0[i] : 0.0f;
  }
  for (int idx = tid; idx < 32 * 4; idx += 64) {
    int r = idx >> 2, c = idx & 3;
    int e = e0 + r;
    float v = 0.0f;
    if (e < NE && c < 3) {
      float a0 = ea[2 * e], a1 = ea[2 * e + 1];
      float eav = (c == 0) ? a0 : (c == 1) ? a1 : sqrtf(a0 * a0 + a1 * a1);
      v = (eav - norm[2 + c]) * norm[10 + c];
    }
    inbuf[r * 8 + c] = v;
  }
  __syncthreads();
  int wv = tid >> 5, lane = tid & 31, nlane = lane & 15, hi = lane >> 4;
  v8f acc[4];
  wmma_layer(sw0, B0, inbuf + (wv * 16 + nlane) * 8, 4, acc, nlane, hi);
  store_acc(act + (wv * 16) * S_ACT, acc, nlane, hi, true);
  hidden_tail_ln(act, mrow, rrow, W1, B1, W2, B2);
  for (int idx = tid; idx < 32 * 64; idx += 64) {
    int r = idx >> 6, c = idx & 63;
    int e = e0 + r;
    if (e < NE)
      eal[e * 64 + c] = (act[r * S_ACT + c] - mrow[r]) * rrow[r] * lng[c] + lnb[c];
  }
}

// ---------------------------------------------------------------------------
// Processor edge MLP: [xl[row],xl[col],eal] (K=192) -> ... + LN = e_new;
// fused: eal += e_new, s[col] += e_new (f32 atomics)
// ---------------------------------------------------------------------------
__global__ void __launch_bounds__(64)
proc_edge_kernel(const int* __restrict__ ei, const float* __restrict__ xl,
                 float* __restrict__ eal, float* __restrict__ sbuf,
                 const float* W0, const float* B0, const float* W1, const float* B1,
                 const float* W2, const float* B2,
                 const float* lng, const float* lnb)
{
  const int KP = 196;
  __shared__ float inbuf[32 * 196];
  __shared__ float act[32 * S_ACT];
  __shared__ int srow[32], scol[32];
  __shared__ float mrow[32], rrow[32];
  int e0 = blockIdx.x * 32;
  int tid = threadIdx.x;
  if (tid < 32) {
    int e = e0 + tid;
    srow[tid] = (e < NE) ? ei[e] : 0;
    scol[tid] = (e < NE) ? ei[NE + e] : 0;
  }
  __syncthreads();
  for (int idx = tid; idx < 32 * 192; idx += 64) {
    int r = idx / 192, k = idx - r * 192;
    int e = e0 + r;
    float v = 0.0f;
    if (e < NE) {
      if (k < 64)       v = xl[srow[r] * 64 + k];
      else if (k < 128) v = xl[scol[r] * 64 + (k - 64)];
      else              v = eal[e * 64 + (k - 128)];
    }
    inbuf[r * KP + k] = v;
  }
  __syncthreads();
  int wv = tid >> 5, lane = tid & 31, nlane = lane & 15, hi = lane >> 4;
  v8f acc[4];
  wmma_layer(W0, B0, inbuf + (wv * 16 + nlane) * KP, 192, acc, nlane, hi);
  store_acc(act + (wv * 16) * S_ACT, acc, nlane, hi, true);
  hidden_tail_ln(act, mrow, rrow, W1, B1, W2, B2);
  for (int idx = tid; idx < 32 * 64; idx += 64) {
    int r = idx >> 6, c = idx & 63;
    int e = e0 + r;
    if (e < NE) {
      float en = (act[r * S_ACT + c] - mrow[r]) * rrow[r] * lng[c] + lnb[c];
      eal[e * 64 + c] = inbuf[r * KP + 128 + c] + en;      // eal += e_new
      atomicAdd(&sbuf[scol[r] * 64 + c], en);              // segment_sum
    }
  }
}

// ---------------------------------------------------------------------------
// Processor node MLP: [xl, s] (K=128) -> ... + LN = x_new; fused xl += x_new
// ---------------------------------------------------------------------------
__global__ void __launch_bounds__(64)
proc_node_kernel(float* __restrict__ xl, const float* __restrict__ sbuf,
                 const float* W0, const float* B0, const float* W1, const float* B1,
                 const float* W2, const float* B2,
                 const float* lng, const float* lnb)
{
  const int KP = 132;
  __shared__ float inbuf[32 * 132];
  __shared__ float act[32 * S_ACT];
  __shared__ float mrow[32], rrow[32];
  int n0 = blockIdx.x * 32;
  int tid = threadIdx.x;
  for (int idx = tid; idx < 32 * 128; idx += 64) {
    int r = idx >> 7, c = idx & 127;
    int n = n0 + r;
    float v = 0.0f;
    if (n < NN) v = (c < 64) ? xl[n * 64 + c] : sbuf[n * 64 + (c - 64)];
    inbuf[r * KP + c] = v;
  }
  __syncthreads();
  int wv = tid >> 5, lane = tid & 31, nlane = lane & 15, hi = lane >> 4;
  v8f acc[4];
  wmma_layer(W0, B0, inbuf + (wv * 16 + nlane) * KP, 128, acc, nlane, hi);
  store_acc(act + (wv * 16) * S_ACT, acc, nlane, hi, true);
  hidden_tail_ln(act, mrow, rrow, W1, B1, W2, B2);
  for (int idx = tid; idx < 32 * 64; idx += 64) {
    int r = idx >> 6, c = idx & 63;
    int n = n0 + r;
    if (n < NN) {
      float xnv = (act[r * S_ACT + c] - mrow[r]) * rrow[r] * lng[c] + lnb[c];
      xl[n * 64 + c] = inbuf[r * KP + c] + xnv;            // xl += x_new
    }
  }
}

// ---------------------------------------------------------------------------
// Decoder: [lc, xl] (K=66, padded to 68) -> 64 -> 64 -> 1 (no LN)
// ---------------------------------------------------------------------------
__global__ void __launch_bounds__(64)
decode_kernel(const float* __restrict__ lc, const float* __restrict__ xl,
              const float* W0, const float* B0, const float* W1, const float* B1,
              const float* W2, const float* B2, float* __restrict__ out)
{
  const int KP = 72;
  __shared__ float sw0[68 * 64];    // W0 zero-padded K=66 -> 68 (17.4 KB)
  __shared__ float inbuf[32 * 72];
  __shared__ float act[32 * S_ACT];
  int n0 = blockIdx.x * 32;
  int tid = threadIdx.x;
  for (int i = tid; i < 68 * 64; i += 64) {
    int k = i >> 6;
    sw0[i] = (k < 66) ? W0[i] : 0.0f;
  }
  for (int idx = tid; idx < 32 * 68; idx += 64) {
    int r = idx / 68, k = idx - r * 68;
    int n = n0 + r;
    float v = 0.0f;
    if (n < NN && k < 66) v = (k < 2) ? lc[n * 2 + k] : xl[n * 64 + (k - 2)];
    inbuf[r * KP + k] = v;
  }
  __syncthreads();
  int wv = tid >> 5, lane = tid & 31, nlane = lane & 15, hi = lane >> 4;
  const float* Arow2 = act + (wv * 16 + nlane) * S_ACT;
  float* actw = act + (wv * 16) * S_ACT;
  v8f acc[4];
  wmma_layer(sw0, B0, inbuf + (wv * 16 + nlane) * KP, 68, acc, nlane, hi);
  store_acc(actw, acc, nlane, hi, true);
  wmma_layer(W1, B1, Arow2, 64, acc, nlane, hi);
  store_acc(actw, acc, nlane, hi, true);
  __syncthreads();
  if (tid < 32) {
    int n = n0 + tid;
    if (n < NN) {
      const float* ar = act + tid * S_ACT;
      float o = B2[0];
#pragma unroll
      for (int k = 0; k < 64; ++k) o += ar[k] * W2[k];
      out[n] = o;
    }
  }
}

// ---------------------------------------------------------------------------
// Host-side launch. Input flattening assumption: top-level dict in insertion
// order; each MLP param dict flattened by sorted keys: bs[0..2], (ln_b, ln_g),
// ws[0..2].
// ---------------------------------------------------------------------------
extern "C" void kernel_launch(void* const* d_in, const int* in_sizes, int n_in,
                              void* d_out, int out_size, void* d_ws, size_t ws_size,
                              hipStream_t stream)
{
  (void)in_sizes; (void)n_in; (void)out_size; (void)ws_size;
  const float* x     = (const float*)d_in[0];
  const int*   ei    = (const int*)d_in[1];
  const float* eattr = (const float*)d_in[2];
  // light_p: bs0,bs1,bs2, ws0,ws1,ws2
  const float* l_b0 = (const float*)d_in[3];
  const float* l_b1 = (const float*)d_in[4];
  const float* l_b2 = (const float*)d_in[5];
  const float* l_w0 = (const float*)d_in[6];
  const float* l_w1 = (const float*)d_in[7];
  const float* l_w2 = (const float*)d_in[8];
  // enc_node_p
  const float* en_b0 = (const float*)d_in[9];
  const float* en_b1 = (const float*)d_in[10];
  const float* en_b2 = (const float*)d_in[11];
  const float* en_lnb = (const float*)d_in[12];
  const float* en_lng = (const float*)d_in[13];
  const float* en_w0 = (const float*)d_in[14];
  const float* en_w1 = (const float*)d_in[15];
  const float* en_w2 = (const float*)d_in[16];
  // enc_edge_p
  const float* ee_b0 = (const float*)d_in[17];
  const float* ee_b1 = (const float*)d_in[18];
  const float* ee_b2 = (const float*)d_in[19];
  const float* ee_lnb = (const float*)d_in[20];
  const float* ee_lng = (const float*)d_in[21];
  const float* ee_w0 = (const float*)d_in[22];
  const float* ee_w1 = (const float*)d_in[23];
  const float* ee_w2 = (const float*)d_in[24];
  // proc_edge_p
  const float* pe_b0 = (const float*)d_in[25];
  const float* pe_b1 = (const float*)d_in[26];
  const float* pe_b2 = (const float*)d_in[27];
  const float* pe_lnb = (const float*)d_in[28];
  const float* pe_lng = (const float*)d_in[29];
  const float* pe_w0 = (const float*)d_in[30];
  const float* pe_w1 = (const float*)d_in[31];
  const float* pe_w2 = (const float*)d_in[32];
  // proc_node_p
  const float* pn_b0 = (const float*)d_in[33];
  const float* pn_b1 = (const float*)d_in[34];
  const float* pn_b2 = (const float*)d_in[35];
  const float* pn_lnb = (const float*)d_in[36];
  const float* pn_lng = (const float*)d_in[37];
  const float* pn_w0 = (const float*)d_in[38];
  const float* pn_w1 = (const float*)d_in[39];
  const float* pn_w2 = (const float*)d_in[40];
  // dec_p
  const float* de_b0 = (const float*)d_in[41];
  const float* de_b1 = (const float*)d_in[42];
  const float* de_b2 = (const float*)d_in[43];
  const float* de_w0 = (const float*)d_in[44];
  const float* de_w1 = (const float*)d_in[45];
  const float* de_w2 = (const float*)d_in[46];

  // Workspace carve-out (floats): ~231 MB total
  float* ws   = (float*)d_ws;
  float* acc  = ws;                 // 16
  float* norm = ws + 16;            // 16
  float* lc   = ws + 32;            // 100000  (50000 x 2)
  float* xl   = lc + 100000;        // 3,200,000  (50000 x 64)
  float* eal  = xl + 3200000;       // 51,200,000 (800000 x 64)
  float* sbuf = eal + 51200000;     // 3,200,000  (50000 x 64)
  float* out  = (float*)d_out;

  // stats + lc accumulators to zero
  zero_kernel<<<(100032 + 255) / 256, 256, 0, stream>>>(ws, 100032);
  node_stats_kernel<<<(NN + 255) / 256, 256, 0, stream>>>(x, acc);
  edge_stats_kernel<<<(NE + 255) / 256, 256, 0, stream>>>(eattr, acc);
  finalize_stats_kernel<<<1, 32, 0, stream>>>(acc, norm);

  light_edge_kernel<<<(NE + 255) / 256, 256, 0, stream>>>(
      x, ei, eattr, l_w0, l_b0, l_w1, l_b1, l_w2, l_b2, lc);

  enc_node_kernel<<<(NN + 31) / 32, 64, 0, stream>>>(
      x, norm, en_w0, en_b0, en_w1, en_b1, en_w2, en_b2, en_lng, en_lnb, xl);
  enc_edge_kernel<<<(NE + 31) / 32, 64, 0, stream>>>(
      eattr, norm, ee_w0, ee_b0, ee_w1, ee_b1, ee_w2, ee_b2, ee_lng, ee_lnb, eal);

  for (int l = 0; l < 4; ++l) {
    zero_kernel<<<(3200000 + 255) / 256, 256, 0, stream>>>(sbuf, 3200000);
    proc_edge_kernel<<<(NE + 31) / 32, 64, 0, stream>>>(
        ei, xl, eal, sbuf, pe_w0, pe_b0, pe_w1, pe_b1, pe_w2, pe_b2, pe_lng, pe_lnb);
    proc_node_kernel<<<(NN + 31) / 32, 64, 0, stream>>>(
        xl, sbuf, pn_w0, pn_b0, pn_w1, pn_b1, pn_w2, pn_b2, pn_lng, pn_lnb);
  }

  decode_kernel<<<(NN + 31) / 32, 64, 0, stream>>>(
      lc, xl, de_w0, de_b0, de_w1, de_b1, de_w2, de_b2, out);
}